// SAGE_49632642073099
// MI455X (gfx1250) — compile-verified
//
#include <hip/hip_runtime.h>
#include <math.h>

typedef __attribute__((ext_vector_type(2))) float v2f;
typedef __attribute__((ext_vector_type(8))) float v8f;

#define DHID 128
#define DPAD 48   // final layer 40 -> padded to 48 (3 col tiles, guard-free loads)

// ---------------------------------------------------------------- utilities
__global__ void k_fill0(float* __restrict__ p, int n) {
  int i = blockIdx.x * blockDim.x + threadIdx.x;
  int stride = gridDim.x * blockDim.x;
  for (; i < n; i += stride) p[i] = 0.0f;
}

// pad a [K x D] row-major matrix to [K x Dp], zero-filling the new columns
__global__ void k_pad(const float* __restrict__ W, float* __restrict__ Wp,
                      int K, int D, int Dp) {
  int i = blockIdx.x * blockDim.x + threadIdx.x;
  if (i < K * Dp) {
    int k = i / Dp, c = i - k * Dp;
    Wp[i] = (c < D) ? W[k * D + c] : 0.0f;
  }
}

__global__ void k_degree(const int* __restrict__ dst, float* __restrict__ deg, int nE) {
  int i = blockIdx.x * blockDim.x + threadIdx.x;
  int stride = gridDim.x * blockDim.x;
  for (; i < nE; i += stride) unsafeAtomicAdd(&deg[dst[i]], 1.0f);
}

__global__ void k_invdeg(float* __restrict__ deg, int n) {
  int i = blockIdx.x * blockDim.x + threadIdx.x;
  if (i < n) deg[i] = 1.0f / fmaxf(deg[i], 1.0f);
}

// --------------------------------------------------- edge gather+segment-sum
// One wave per edge (grid-stride): 128 channels = float4 per lane,
// 4 global_atomic_add_f32 per lane. Entire h (10MB) is L2-resident.
__global__ void k_aggregate(const float* __restrict__ h, const int* __restrict__ src,
                            const int* __restrict__ dst, float* __restrict__ agg,
                            int nE, int nWaves) {
  const int lane = threadIdx.x & 31;
  int w = blockIdx.x * (blockDim.x >> 5) + (threadIdx.x >> 5);
  for (int e = w; e < nE; e += nWaves) {
    const int s = src[e];
    const int d = dst[e];
    const float4 v = *reinterpret_cast<const float4*>(h + (size_t)s * DHID + (lane << 2));
    float* p = agg + (size_t)d * DHID + (lane << 2);
    unsafeAtomicAdd(p + 0, v.x);
    unsafeAtomicAdd(p + 1, v.y);
    unsafeAtomicAdd(p + 2, v.z);
    unsafeAtomicAdd(p + 3, v.w);
  }
}

// ------------------------------------------------------ fused SAGE GEMM (WMMA)
// out[N x D] = act( A1 @ W1 + (A2 * invdeg_row) @ W2 + bias ), K == 128.
// D is compile-time (128 or 48) -> unconditional B loads, no exec divergence.
// One wave computes one 16x16 C tile with V_WMMA_F32_16X16X4_F32; two
// independent accumulator chains (self / neigh) interleave in one K-loop so
// the matrix pipe sees 2x ILP instead of one serial RAW chain.
// A layout (16x4): lane<16 holds K={k0,k0+1}, lane>=16 holds K={k0+2,k0+3}.
// B layout (4x16): mirrored across half-waves; C: VGPR r -> row m0+r+8*half.
template <int D, bool RELU>
__global__ void k_sage_gemm(const float* __restrict__ A1, const float* __restrict__ A2,
                            const float* __restrict__ invdeg,
                            const float* __restrict__ W1, const float* __restrict__ W2,
                            const float* __restrict__ bias, float* __restrict__ out,
                            int nRows, int DoutStore) {
  constexpr int NCT = D / 16;
  const int lane  = threadIdx.x & 31;
  const int gwave = blockIdx.x * (blockDim.x >> 5) + (threadIdx.x >> 5);
  const int totalTiles = (nRows >> 4) * NCT;
  if (gwave >= totalTiles) return;           // wave-uniform exit: EXEC stays full
  const int tr = gwave / NCT;
  const int tc = gwave - tr * NCT;
  const int m0 = tr << 4;
  const int n0 = tc << 4;
  const int half = lane >> 4;
  const int lr   = lane & 15;
  const int row  = m0 + lr;                  // A row this lane feeds
  const int col  = n0 + lr;                  // B/out column this lane feeds

  const float idg = invdeg[row];
  const float* aRow1 = A1 + (size_t)row * DHID + 2 * half;
  const float* aRow2 = A2 + (size_t)row * DHID + 2 * half;
  const float* bCol1 = W1 + 2 * half * D + col;
  const float* bCol2 = W2 + 2 * half * D + col;

  v8f acc1 = {};
  v8f acc2 = {};

  #pragma unroll 4
  for (int k0 = 0; k0 < DHID; k0 += 4) {
    v2f a1; a1.x = aRow1[k0];       a1.y = aRow1[k0 + 1];
    v2f a2; a2.x = aRow2[k0] * idg; a2.y = aRow2[k0 + 1] * idg;
    v2f b1; b1.x = bCol1[k0 * D];   b1.y = bCol1[(k0 + 1) * D];
    v2f b2; b2.x = bCol2[k0 * D];   b2.y = bCol2[(k0 + 1) * D];
    acc1 = __builtin_amdgcn_wmma_f32_16x16x4_f32(false, a1, false, b1,
                                                 (short)0, acc1, false, false);
    acc2 = __builtin_amdgcn_wmma_f32_16x16x4_f32(false, a2, false, b2,
                                                 (short)0, acc2, false, false);
  }

  if (col < DoutStore) {
    const float bv = bias[col];
    #pragma unroll
    for (int r = 0; r < 8; ++r) {
      float v = acc1[r] + acc2[r] + bv;
      if (RELU) v = fmaxf(v, 0.0f);
      out[(size_t)(m0 + r + 8 * half) * DoutStore + col] = v;
    }
  }
}

// ------------------------------------------------------------- BatchNorm
__global__ void k_bn_stats(const float* __restrict__ h, float* __restrict__ stats,
                           int nRows) {
  const int c = threadIdx.x;                 // blockDim.x == 128 channels
  float s = 0.0f, s2 = 0.0f;
  for (int i = blockIdx.x; i < nRows; i += gridDim.x) {
    const float v = h[(size_t)i * DHID + c];
    s += v; s2 += v * v;
  }
  unsafeAtomicAdd(&stats[c], s);
  unsafeAtomicAdd(&stats[DHID + c], s2);
}

__global__ void k_bn_finalize(const float* __restrict__ stats,
                              const float* __restrict__ gamma,
                              const float* __restrict__ beta,
                              float* __restrict__ ss, int nRows) {
  const int c = threadIdx.x;
  if (c < DHID) {
    const float invN = 1.0f / (float)nRows;
    const float mean = stats[c] * invN;
    const float var  = stats[DHID + c] * invN - mean * mean;
    const float sc   = gamma[c] * rsqrtf(var + 1e-5f);
    ss[c]        = sc;
    ss[DHID + c] = beta[c] - mean * sc;
  }
}

__global__ void k_bn_apply(const float* __restrict__ hin, const float* __restrict__ ss,
                           float* __restrict__ hout, int nVec4) {
  const int i = blockIdx.x * blockDim.x + threadIdx.x;
  if (i < nVec4) {
    const int idx = i << 2;
    const int c   = idx & (DHID - 1);        // DHID multiple of 4 -> same row
    const float4 v = *reinterpret_cast<const float4*>(hin + idx);
    float4 o;
    o.x = v.x * ss[c + 0] + ss[DHID + c + 0];
    o.y = v.y * ss[c + 1] + ss[DHID + c + 1];
    o.z = v.z * ss[c + 2] + ss[DHID + c + 2];
    o.w = v.w * ss[c + 3] + ss[DHID + c + 3];
    *reinterpret_cast<float4*>(hout + idx) = o;
  }
}

// ------------------------------------------------------------- log_softmax(40)
__global__ void k_log_softmax40(float* __restrict__ out, int nRows) {
  const int lane = threadIdx.x & 31;
  const int row  = blockIdx.x * (blockDim.x >> 5) + (threadIdx.x >> 5);
  if (row >= nRows) return;
  float* p = out + (size_t)row * 40;
  const float a = p[lane];                                  // cols 0..31
  const float b = (lane < 8) ? p[lane + 32] : -__builtin_inff();
  float m = fmaxf(a, b);
  #pragma unroll
  for (int off = 16; off; off >>= 1) m = fmaxf(m, __shfl_xor(m, off, 32));
  float e = expf(a - m) + ((lane < 8) ? expf(b - m) : 0.0f);
  #pragma unroll
  for (int off = 16; off; off >>= 1) e += __shfl_xor(e, off, 32);
  const float ls = m + logf(e);
  p[lane] = a - ls;
  if (lane < 8) p[lane + 32] = b - ls;
}

// ---------------------------------------------------------------- launcher
extern "C" void kernel_launch(void* const* d_in, const int* in_sizes, int n_in,
                              void* d_out, int out_size, void* d_ws, size_t ws_size,
                              hipStream_t stream) {
  const float* x    = (const float*)d_in[0];
  const int*   src  = (const int*)  d_in[1];
  const int*   dst  = (const int*)  d_in[2];
  const float* ws0  = (const float*)d_in[3];
  const float* wn0  = (const float*)d_in[4];
  const float* b0   = (const float*)d_in[5];
  const float* g0   = (const float*)d_in[6];
  const float* bt0  = (const float*)d_in[7];
  const float* ws1  = (const float*)d_in[8];
  const float* wn1  = (const float*)d_in[9];
  const float* b1   = (const float*)d_in[10];
  const float* g1   = (const float*)d_in[11];
  const float* bt1  = (const float*)d_in[12];
  const float* ws2  = (const float*)d_in[13];
  const float* wn2  = (const float*)d_in[14];
  const float* b2   = (const float*)d_in[15];
  float* out = (float*)d_out;

  const int N    = in_sizes[0] / DHID;   // 20000
  const int E    = in_sizes[1];          // 640000
  const int DOUT = in_sizes[15];         // 40

  // scratch layout (floats)
  float* wsf   = (float*)d_ws;
  float* deg   = wsf;                         // N          (becomes invdeg)
  float* stats = wsf + N;                     // 2*DHID sums
  float* ssbuf = stats + 2 * DHID;            // 2*DHID scale/shift
  float* wpS   = ssbuf + 2 * DHID;            // DHID*DPAD  padded ws2
  float* wpN   = wpS + DHID * DPAD;           // DHID*DPAD  padded wn2
  float* bp    = wpN + DHID * DPAD;           // DPAD       padded b2
  float* agg   = bp + DPAD;                   // N*DHID
  float* hb    = agg + (size_t)N * DHID;      // N*DHID (pre-BN)
  float* ha    = hb  + (size_t)N * DHID;      // N*DHID (post-BN)

  const int HW = N * DHID;                    // 2.56M floats

  const dim3 blk(256);
  const int aggBlocks  = 2048;                 // 8 waves/block
  const int aggWaves   = aggBlocks * 8;
  const int tiles128   = (N / 16) * (DHID / 16);
  const int gemmB128   = (tiles128 + 7) / 8;
  const int tiles48    = (N / 16) * (DPAD / 16);
  const int gemmB48    = (tiles48 + 7) / 8;

  // degrees + padded final-layer weights (recomputed every call: ws is scratch)
  k_fill0<<<256, blk, 0, stream>>>(deg, N);
  k_degree<<<1024, blk, 0, stream>>>(dst, deg, E);
  k_invdeg<<<(N + 255) / 256, blk, 0, stream>>>(deg, N);
  k_pad<<<(DHID * DPAD + 255) / 256, blk, 0, stream>>>(ws2, wpS, DHID, DOUT, DPAD);
  k_pad<<<(DHID * DPAD + 255) / 256, blk, 0, stream>>>(wn2, wpN, DHID, DOUT, DPAD);
  k_pad<<<1, blk, 0, stream>>>(b2, bp, 1, DOUT, DPAD);

  // ---------------- layer 0: x -> hb -> BN -> ha
  k_fill0<<<2048, blk, 0, stream>>>(agg, HW);
  k_fill0<<<1, blk, 0, stream>>>(stats, 2 * DHID);
  k_aggregate<<<aggBlocks, blk, 0, stream>>>(x, src, dst, agg, E, aggWaves);
  k_sage_gemm<DHID, true><<<gemmB128, blk, 0, stream>>>(x, agg, deg, ws0, wn0, b0,
                                                        hb, N, DHID);
  k_bn_stats<<<250, DHID, 0, stream>>>(hb, stats, N);
  k_bn_finalize<<<1, DHID, 0, stream>>>(stats, g0, bt0, ssbuf, N);
  k_bn_apply<<<(HW / 4 + 255) / 256, blk, 0, stream>>>(hb, ssbuf, ha, HW / 4);

  // ---------------- layer 1: ha -> hb -> BN -> ha
  k_fill0<<<2048, blk, 0, stream>>>(agg, HW);
  k_fill0<<<1, blk, 0, stream>>>(stats, 2 * DHID);
  k_aggregate<<<aggBlocks, blk, 0, stream>>>(ha, src, dst, agg, E, aggWaves);
  k_sage_gemm<DHID, true><<<gemmB128, blk, 0, stream>>>(ha, agg, deg, ws1, wn1, b1,
                                                        hb, N, DHID);
  k_bn_stats<<<250, DHID, 0, stream>>>(hb, stats, N);
  k_bn_finalize<<<1, DHID, 0, stream>>>(stats, g1, bt1, ssbuf, N);
  k_bn_apply<<<(HW / 4 + 255) / 256, blk, 0, stream>>>(hb, ssbuf, ha, HW / 4);

  // ---------------- layer 2: ha -> d_out -> log_softmax (in place)
  k_fill0<<<2048, blk, 0, stream>>>(agg, HW);
  k_aggregate<<<aggBlocks, blk, 0, stream>>>(ha, src, dst, agg, E, aggWaves);
  k_sage_gemm<DPAD, false><<<gemmB48, blk, 0, stream>>>(ha, agg, deg, wpS, wpN, bp,
                                                        out, N, DOUT);
  k_log_softmax40<<<(N + 7) / 8, blk, 0, stream>>>(out, N);
}